// Model_25709674234168
// MI455X (gfx1250) — compile-verified
//
#include <hip/hip_runtime.h>

typedef __attribute__((ext_vector_type(2))) float v2f;
typedef __attribute__((ext_vector_type(8))) float v8f;

#define EDGE_EPS 1e-8f
#define BLOCK 128

struct F3 { float x, y, z; };

__global__ __launch_bounds__(BLOCK) void edge_feature_kernel(
    const float* __restrict__ verts,        // [NV,3]
    const int*   __restrict__ faces,        // [NF,3]
    const int*   __restrict__ nodes_from,   // [NE]
    const int*   __restrict__ faces_to,     // [NE]
    const int*   __restrict__ edge_labels,  // [NE]
    const float* __restrict__ edge_emb,     // [4,4]
    float*       __restrict__ out,          // [NE,9]
    int n_edges)
{
    __shared__ float stage[BLOCK * 9];      // row-major [localEdge][9] staging

    const int tid  = threadIdx.x;
    const int lane = tid & 31;
    const int wave = tid >> 5;
    const int blockBase = (int)blockIdx.x * BLOCK;
    // Uniform: does this block own a full set of BLOCK valid edges?
    const bool fullBlock = (blockBase + BLOCK) <= n_edges;

    const int e  = blockBase + tid;
    const int ec = min(e, n_edges - 1);     // no-op select for all but the tail block

    // ---------------- geometry (gather + cross + project) ----------------
    const int f  = faces_to[ec];
    const int nf = nodes_from[ec];
    const int i0 = faces[3 * f + 0];
    const int i1 = faces[3 * f + 1];
    const int i2 = faces[3 * f + 2];

    const F3 v0 = *(const F3*)(verts + 3 * i0);
    const F3 v1 = *(const F3*)(verts + 3 * i1);
    const F3 v2 = *(const F3*)(verts + 3 * i2);
    const F3 pf = *(const F3*)(verts + 3 * nf);

    const float ax = v1.x - v0.x, ay = v1.y - v0.y, az = v1.z - v0.z;
    const float bx = v2.x - v0.x, by = v2.y - v0.y, bz = v2.z - v0.z;
    float nx = ay * bz - az * by;
    float ny = az * bx - ax * bz;
    float nz = ax * by - ay * bx;
    const float inv = 1.0f / (sqrtf(nx * nx + ny * ny + nz * nz) + EDGE_EPS);
    nx *= inv; ny *= inv; nz *= inv;

    const float rx = pf.x - v0.x, ry = pf.y - v0.y, rz = pf.z - v0.z;
    const float d  = rx * nx + ry * ny + rz * nz;
    const float ad = fabsf(d);
    // where(d<0, -(d*n), d*n) == |d| * n  componentwise

    stage[tid * 9 + 0] = ad * nx;
    stage[tid * 9 + 1] = ad * ny;
    stage[tid * 9 + 2] = ad * nz;
    stage[tid * 9 + 3] = ad;
    stage[tid * 9 + 4] = 1.0f;

    // ---------------- embedding via V_WMMA_F32_16X16X4_F32 ----------------
    // D = onehot(labels) (16x4, f32, exact) x edge_emb padded to 4x16.
    // A layout (16x4 f32): VGPR0 = {K=0 | K=2}, VGPR1 = {K=1 | K=3} across lane halves
    // B layout (4x16 f32): same K-split, N = lane & 15
    const int col = lane & 15;            // N (and M for A) index
    const int kr  = (lane < 16) ? 0 : 2;  // K rows held by this lane half
    const int cc  = (col < 4) ? col : 0;
    v2f bmat;
    bmat.x = (col < 4) ? edge_emb[kr * 4 + cc]       : 0.0f;
    bmat.y = (col < 4) ? edge_emb[(kr + 1) * 4 + cc] : 0.0f;

    const int waveBase = blockBase + wave * 32;

#pragma unroll
    for (int h = 0; h < 2; ++h) {
        const int me  = min(waveBase + h * 16 + col, n_edges - 1);
        const int lbl = edge_labels[me];

        v2f amat;
        amat.x = (lbl == kr)     ? 1.0f : 0.0f;
        amat.y = (lbl == kr + 1) ? 1.0f : 0.0f;

        v8f cz = {};
        v8f dmat = __builtin_amdgcn_wmma_f32_16x16x4_f32(
            /*neg_a=*/false, amat, /*neg_b=*/false, bmat,
            /*c_mod=*/(short)0, cz, /*reuse_a=*/false, /*reuse_b=*/false);

        // D layout: VGPR r, lanes 0-15 -> (M=r, N=lane); lanes 16-31 -> (M=r+8, N=lane-16)
        if (col < 4) {
            const int mbase = (lane < 16) ? 0 : 8;
            const int le0 = wave * 32 + h * 16;
#pragma unroll
            for (int r = 0; r < 8; ++r) {
                stage[(le0 + mbase + r) * 9 + 5 + col] = dmat[r];
            }
        }
    }

    __syncthreads();

    // ---------------- coalesced writeback: 1152 contiguous floats ----------------
    const int outBase = blockBase * 9;
    if (fullBlock) {
        // Uniform fast path: no predication, clean ds_load/global_store clause.
#pragma unroll
        for (int k = 0; k < 9; ++k) {
            out[outBase + k * BLOCK + tid] = stage[k * BLOCK + tid];
        }
    } else {
        const int total = n_edges * 9;
#pragma unroll
        for (int k = 0; k < 9; ++k) {
            const int idx = outBase + k * BLOCK + tid;
            if (idx < total) out[idx] = stage[k * BLOCK + tid];
        }
    }
}

extern "C" void kernel_launch(void* const* d_in, const int* in_sizes, int n_in,
                              void* d_out, int out_size, void* d_ws, size_t ws_size,
                              hipStream_t stream) {
    const float* verts       = (const float*)d_in[0];
    // d_in[1] = icontour_grad : unused by the reference output
    const int*   faces       = (const int*)d_in[2];
    const int*   nodes_from  = (const int*)d_in[3];
    const int*   faces_to    = (const int*)d_in[4];
    const int*   edge_labels = (const int*)d_in[5];
    const float* edge_emb    = (const float*)d_in[6];
    float*       out         = (float*)d_out;

    const int n_edges = in_sizes[3];
    const int grid = (n_edges + BLOCK - 1) / BLOCK;

    edge_feature_kernel<<<grid, BLOCK, 0, stream>>>(
        verts, faces, nodes_from, faces_to, edge_labels, edge_emb, out, n_edges);
}